// TensorProductExpansion_28432683499862
// MI455X (gfx1250) — compile-verified
//
#include <hip/hip_runtime.h>
#include <hip/hip_bf16.h>

typedef __attribute__((ext_vector_type(2))) float v2f;
typedef __attribute__((ext_vector_type(8))) float v8f;

// D[16x25] = A[16x5] x B[5x25] per 16-row tile, via V_WMMA_F32_16X16X4_F32.
// K=5 -> two K=4 chunks (chunk1 = {k4,0,0,0}); N=25 -> two 16-wide N tiles.
// Each wave-iteration handles 32 rows = two row-tiles for extra MLP.
__global__ void __launch_bounds__(256)
tpe_wmma_kernel(const float* __restrict__ feat, const float* __restrict__ cg,
                float* __restrict__ out, int ntiles32) {
    const int lane = threadIdx.x & 31;
    const int wave = threadIdx.x >> 5;
    const int gwave = blockIdx.x * (blockDim.x >> 5) + wave;
    const int nwaves = gridDim.x * (blockDim.x >> 5);

    const int n0   = lane & 15;   // N column (B/C/D) and M row (A) within tile
    const int half = lane >> 4;   // K-pair selector for A/B register striping

    // ---- Build B matrices once (cg is tiny & constant): B[k][n] = cg[n*5 + k]
    // B 4x16 f32 layout: lane n (0..15) col n, vgpr0 = K=2*half row, vgpr1 = K=2*half+1.
    v2f blo0, blo1, bhi0, bhi1;
    {
        const int n = n0;                 // N-tile 0: cols 0..15
        blo0.x = cg[n * 5 + 2 * half];
        blo0.y = cg[n * 5 + 2 * half + 1];
        bhi0.x = (half == 0) ? cg[n * 5 + 4] : 0.0f;   // K=4 lives in slot K'=0
        bhi0.y = 0.0f;
        const int n1 = n0 + 16;           // N-tile 1: cols 16..24 valid
        if (n1 < 25) {
            blo1.x = cg[n1 * 5 + 2 * half];
            blo1.y = cg[n1 * 5 + 2 * half + 1];
            bhi1.x = (half == 0) ? cg[n1 * 5 + 4] : 0.0f;
        } else {
            blo1.x = 0.0f; blo1.y = 0.0f; bhi1.x = 0.0f;
        }
        bhi1.y = 0.0f;
    }

    for (int t = gwave; t < ntiles32; t += nwaves) {
        const long base = (long)t * 32;
        // A 16x4 f32 layout: lane M=n0; vgpr0 = K=2*half, vgpr1 = K=2*half+1
        const float* fp0 = feat + (base + n0) * 5;
        const float* fp1 = feat + (base + 16 + n0) * 5;
        // Unconditional loads (all addresses in bounds); zero via select -> no EXEC branch.
        v2f alo0, alo1;
        alo0.x = fp0[2 * half];
        alo0.y = fp0[2 * half + 1];
        alo1.x = fp1[2 * half];
        alo1.y = fp1[2 * half + 1];
        const float k4a = fp0[4];
        const float k4b = fp1[4];
        v2f ahi0, ahi1;
        ahi0.x = (half == 0) ? k4a : 0.0f;  // K-chunk 1: only K=4 nonzero
        ahi0.y = 0.0f;
        ahi1.x = (half == 0) ? k4b : 0.0f;
        ahi1.y = 0.0f;

        v8f c00 = {}, c01 = {}, c10 = {}, c11 = {};
        c00 = __builtin_amdgcn_wmma_f32_16x16x4_f32(false, alo0, false, blo0, (short)0, c00, false, false);
        c01 = __builtin_amdgcn_wmma_f32_16x16x4_f32(false, alo0, false, blo1, (short)0, c01, false, false);
        c10 = __builtin_amdgcn_wmma_f32_16x16x4_f32(false, alo1, false, blo0, (short)0, c10, false, false);
        c11 = __builtin_amdgcn_wmma_f32_16x16x4_f32(false, alo1, false, blo1, (short)0, c11, false, false);
        c00 = __builtin_amdgcn_wmma_f32_16x16x4_f32(false, ahi0, false, bhi0, (short)0, c00, false, false);
        c01 = __builtin_amdgcn_wmma_f32_16x16x4_f32(false, ahi0, false, bhi1, (short)0, c01, false, false);
        c10 = __builtin_amdgcn_wmma_f32_16x16x4_f32(false, ahi1, false, bhi0, (short)0, c10, false, false);
        c11 = __builtin_amdgcn_wmma_f32_16x16x4_f32(false, ahi1, false, bhi1, (short)0, c11, false, false);

        // D 16x16 f32 layout: vgpr v, lane L -> row v + 8*half, col (L&15)
        const long rb0 = base + 8 * half;        // row-tile 0
        const long rb1 = base + 16 + 8 * half;   // row-tile 1
        #pragma unroll
        for (int v = 0; v < 8; ++v) {
            __builtin_nontemporal_store(c00[v], &out[(rb0 + v) * 25 + n0]);
        }
        #pragma unroll
        for (int v = 0; v < 8; ++v) {
            __builtin_nontemporal_store(c10[v], &out[(rb1 + v) * 25 + n0]);
        }
        if (n0 < 9) {  // divergence only after all WMMAs; EXEC all-1s at WMMA issue
            #pragma unroll
            for (int v = 0; v < 8; ++v) {
                __builtin_nontemporal_store(c01[v], &out[(rb0 + v) * 25 + 16 + n0]);
            }
            #pragma unroll
            for (int v = 0; v < 8; ++v) {
                __builtin_nontemporal_store(c11[v], &out[(rb1 + v) * 25 + 16 + n0]);
            }
        }
    }
}

// Scalar tail for rows not covered by 32-row super-tiles (not launched for R % 32 == 0).
__global__ void tpe_tail_kernel(const float* __restrict__ feat, const float* __restrict__ cg,
                                float* __restrict__ out, long rowStart, int rem) {
    const int tid = threadIdx.x + blockIdx.x * blockDim.x;
    const int total = rem * 25;
    for (int i = tid; i < total; i += blockDim.x * gridDim.x) {
        const int rr = i / 25, c = i % 25;
        const long r = rowStart + rr;
        const float* fp = feat + r * 5;
        float s = 0.0f;
        #pragma unroll
        for (int k = 0; k < 5; ++k) s += cg[c * 5 + k] * fp[k];
        out[r * 25 + c] = s;
    }
}

extern "C" void kernel_launch(void* const* d_in, const int* in_sizes, int n_in,
                              void* d_out, int out_size, void* d_ws, size_t ws_size,
                              hipStream_t stream) {
    const float* feat = (const float*)d_in[0];   // [R, 5] f32
    const float* cg   = (const float*)d_in[1];   // [5, 5, 5] f32
    float* out = (float*)d_out;                  // [R, 25] f32

    const long R = (long)in_sizes[0] / 5;
    const int ntiles32 = (int)(R / 32);
    const int rem = (int)(R % 32);

    if (ntiles32 > 0) {
        int blocks = (ntiles32 + 7) / 8;         // 8 waves per 256-thread block
        if (blocks > 2048) blocks = 2048;        // ~16K waves, grid-stride over tiles
        tpe_wmma_kernel<<<blocks, 256, 0, stream>>>(feat, cg, out, ntiles32);
    }
    if (rem > 0) {
        tpe_tail_kernel<<<1, 128, 0, stream>>>(feat, cg, out, (long)ntiles32 * 32, rem);
    }
}